// ContGCDEFunc_6794638262307
// MI455X (gfx1250) — compile-verified
//
#include <hip/hip_runtime.h>

typedef float v2f __attribute__((ext_vector_type(2)));
typedef float v8f __attribute__((ext_vector_type(8)));

#define N_FEAT 64  // D_IN == D_HID == 64

// ---------------- utility: zero float4 region ----------------
__global__ void zero_f4(float4* __restrict__ p, int n4) {
    int i = blockIdx.x * blockDim.x + threadIdx.x;
    if (i < n4) p[i] = make_float4(0.f, 0.f, 0.f, 0.f);
}

// ---------------- degree (f32 atomic counts, exact < 2^24) ----------------
__global__ void deg_kernel(const int* __restrict__ dst, float* __restrict__ deg, int n_edges) {
    int e = blockIdx.x * blockDim.x + threadIdx.x;
    if (e < n_edges) atomicAdd(&deg[dst[e]], 1.0f);
}

// ---------------- norm = deg>0 ? deg^-1/2 : 0 (in place) ----------------
__global__ void norm_kernel(float* __restrict__ dn, int n_nodes) {
    int i = blockIdx.x * blockDim.x + threadIdx.x;
    if (i < n_nodes) {
        float d = dn[i];
        dn[i] = (d > 0.f) ? rsqrtf(d) : 0.f;
    }
}

// ---------------- h[n][d] = norm[n] * sum_k x[n][k][chan] * W[k][d] ----------------
// One wave computes a 16x64 output tile: A frag = 16x4 slice of x (strided by
// channel interleave), 4 B frags cover N=64, 16 K-chunks of 4 -> 64 v_wmma.
// ISA f32 16x16x4 layouts: A: lanes0-15 hold K0/K1 (vgpr0/1), lanes16-31 K2/K3.
// C/D: col = lane%16, row = vgprIdx + (lane<16 ? 0 : 8).
__global__ void gemm_norm_wmma(const float* __restrict__ x, int chan,
                               const float* __restrict__ W,
                               const float* __restrict__ norm,
                               float* __restrict__ h, int n_nodes) {
    int wave  = (blockIdx.x * blockDim.x + threadIdx.x) >> 5;
    int lane  = threadIdx.x & 31;
    int node0 = wave * 16;
    if (node0 >= n_nodes) return;           // wave-uniform exit; EXEC stays all-1s
    int half = lane >> 4;                   // 0: K pair {0,1}; 1: K pair {2,3}
    int l16  = lane & 15;

    const float* xrow = x + (size_t)(node0 + l16) * (2 * N_FEAT) + chan; // stride 2 per k
    v8f acc0 = {}, acc1 = {}, acc2 = {}, acc3 = {};

#pragma unroll
    for (int kc = 0; kc < 16; ++kc) {
        int k0 = kc * 4 + half * 2;
        v2f a;
        a.x = xrow[(size_t)(k0 + 0) * 2];
        a.y = xrow[(size_t)(k0 + 1) * 2];
        const float* wr0 = W + (size_t)(k0 + 0) * N_FEAT;
        const float* wr1 = W + (size_t)(k0 + 1) * N_FEAT;
        v2f b0; b0.x = wr0[l16 +  0]; b0.y = wr1[l16 +  0];
        v2f b1; b1.x = wr0[l16 + 16]; b1.y = wr1[l16 + 16];
        v2f b2; b2.x = wr0[l16 + 32]; b2.y = wr1[l16 + 32];
        v2f b3; b3.x = wr0[l16 + 48]; b3.y = wr1[l16 + 48];
        acc0 = __builtin_amdgcn_wmma_f32_16x16x4_f32(false, a, false, b0, (short)0, acc0, false, false);
        acc1 = __builtin_amdgcn_wmma_f32_16x16x4_f32(false, a, false, b1, (short)0, acc1, false, false);
        acc2 = __builtin_amdgcn_wmma_f32_16x16x4_f32(false, a, false, b2, (short)0, acc2, false, false);
        acc3 = __builtin_amdgcn_wmma_f32_16x16x4_f32(false, a, false, b3, (short)0, acc3, false, false);
    }

#pragma unroll
    for (int v = 0; v < 8; ++v) {
        int row = node0 + v + half * 8;
        float nr = norm[row];
        float* hr = h + (size_t)row * N_FEAT + l16;
        hr[ 0] = acc0[v] * nr;
        hr[16] = acc1[v] * nr;
        hr[32] = acc2[v] * nr;
        hr[48] = acc3[v] * nr;
    }
}

// ---------------- scatter-add: agg[dst] += h[src], 16 threads/edge, float4 ----------------
__global__ void scatter_kernel(const float* __restrict__ h,
                               const int* __restrict__ src,
                               const int* __restrict__ dst,
                               float* __restrict__ agg, int n_edges) {
    int t = blockIdx.x * blockDim.x + threadIdx.x;
    int e = t >> 4;
    if (e >= n_edges) return;
    int q = (t & 15) * 4;
    int s = src[e], d = dst[e];
    const float4 v = *(const float4*)(h + (size_t)s * N_FEAT + q);
    float* ap = agg + (size_t)d * N_FEAT + q;
    atomicAdd(ap + 0, v.x);
    atomicAdd(ap + 1, v.y);
    atomicAdd(ap + 2, v.z);
    atomicAdd(ap + 3, v.w);
}

// ---------------- out = relu(agg*norm + b); agg <- 0 for next channel ----------------
__global__ void finalize_first(float* __restrict__ agg, const float* __restrict__ norm,
                               const float* __restrict__ b, float* __restrict__ out,
                               int n_nodes) {
    int t = blockIdx.x * blockDim.x + threadIdx.x;
    if (t >= n_nodes * 16) return;
    int node = t >> 4, q = t & 15;
    float4* ap = (float4*)agg + t;
    float4 a = *ap;
    float nr = norm[node];
    float4 bb = ((const float4*)b)[q];
    float4 r;
    r.x = fmaxf(fmaf(a.x, nr, bb.x), 0.f);
    r.y = fmaxf(fmaf(a.y, nr, bb.y), 0.f);
    r.z = fmaxf(fmaf(a.z, nr, bb.z), 0.f);
    r.w = fmaxf(fmaf(a.w, nr, bb.w), 0.f);
    ((float4*)out)[t] = r;
    *ap = make_float4(0.f, 0.f, 0.f, 0.f);
}

// ---------------- out += relu(agg*norm + b) ----------------
__global__ void finalize_second(const float* __restrict__ agg, const float* __restrict__ norm,
                                const float* __restrict__ b, float* __restrict__ out,
                                int n_nodes) {
    int t = blockIdx.x * blockDim.x + threadIdx.x;
    if (t >= n_nodes * 16) return;
    int node = t >> 4, q = t & 15;
    float4 a = ((const float4*)agg)[t];
    float nr = norm[node];
    float4 bb = ((const float4*)b)[q];
    float4 o = ((float4*)out)[t];
    o.x += fmaxf(fmaf(a.x, nr, bb.x), 0.f);
    o.y += fmaxf(fmaf(a.y, nr, bb.y), 0.f);
    o.z += fmaxf(fmaf(a.z, nr, bb.z), 0.f);
    o.w += fmaxf(fmaf(a.w, nr, bb.w), 0.f);
    ((float4*)out)[t] = o;
}

extern "C" void kernel_launch(void* const* d_in, const int* in_sizes, int n_in,
                              void* d_out, int out_size, void* d_ws, size_t ws_size,
                              hipStream_t stream) {
    // inputs: 0=t(unused), 1=x[n,64,2], 2=src, 3=dst, 4=W_l, 5=b_l, 6=W_c, 7=b_c
    const float* x   = (const float*)d_in[1];
    const int*   src = (const int*)d_in[2];
    const int*   dst = (const int*)d_in[3];
    const float* W_l = (const float*)d_in[4];
    const float* b_l = (const float*)d_in[5];
    const float* W_c = (const float*)d_in[6];
    const float* b_c = (const float*)d_in[7];
    float* out = (float*)d_out;

    const int n_edges = in_sizes[2];
    const int n_nodes = in_sizes[1] / (2 * N_FEAT);

    // workspace layout: [norm | h | agg], ~51.6 MB total
    float* ws   = (float*)d_ws;
    size_t normPad = ((size_t)n_nodes + 127) & ~(size_t)127;   // keep 16B alignment
    float* nrm  = ws;
    float* h    = ws + normPad;
    float* agg  = h + (size_t)n_nodes * N_FEAT;

    // 1) zero deg region + agg
    {
        int n4 = (int)(normPad / 4);
        zero_f4<<<(n4 + 255) / 256, 256, 0, stream>>>((float4*)nrm, n4);
        int a4 = n_nodes * 16;
        zero_f4<<<(a4 + 255) / 256, 256, 0, stream>>>((float4*)agg, a4);
    }
    // 2) degree, 3) norm
    deg_kernel<<<(n_edges + 255) / 256, 256, 0, stream>>>(dst, nrm, n_edges);
    norm_kernel<<<(n_nodes + 255) / 256, 256, 0, stream>>>(nrm, n_nodes);

    int nwaves  = (n_nodes + 15) / 16;
    int gth     = nwaves * 32;
    int sgrid   = (n_edges * 16 + 255) / 256;
    int fgrid   = (n_nodes * 16 + 255) / 256;

    // channel 0 (lower): x[:,:,0] @ W_l
    gemm_norm_wmma<<<(gth + 255) / 256, 256, 0, stream>>>(x, 0, W_l, nrm, h, n_nodes);
    scatter_kernel<<<sgrid, 256, 0, stream>>>(h, src, dst, agg, n_edges);
    finalize_first<<<fgrid, 256, 0, stream>>>(agg, nrm, b_l, out, n_nodes);

    // channel 1 (capital): x[:,:,1] @ W_c
    gemm_norm_wmma<<<(gth + 255) / 256, 256, 0, stream>>>(x, 1, W_c, nrm, h, n_nodes);
    scatter_kernel<<<sgrid, 256, 0, stream>>>(h, src, dst, agg, n_edges);
    finalize_second<<<fgrid, 256, 0, stream>>>(agg, nrm, b_c, out, n_nodes);
}